// LightGCNModel_50422916055128
// MI455X (gfx1250) — compile-verified
//
#include <hip/hip_runtime.h>
#include <stdint.h>

typedef __attribute__((ext_vector_type(2))) float v2f;
typedef __attribute__((ext_vector_type(8))) float v8f;

#define EMBED 64
#define N_LAYERS 3
#define EPW 8   // edges per wave in the scatter kernel

// ---------------------------------------------------------------------------
// Native fp32 scatter-add (no-return form, STOREcnt-tracked). Guarantees
// global_atomic_add_f32 instead of a possible CAS-loop expansion of the
// generic atomic builtin. S_ENDPGM's implicit wait-idle drains STOREcnt.
// ---------------------------------------------------------------------------
__device__ __forceinline__ void atomic_add_f32(float* p, float v) {
  asm volatile("global_atomic_add_f32 %0, %1, off"
               :
               : "v"(p), "v"(v)
               : "memory");
}

// ---------------------------------------------------------------------------
// e0 = concat(user_embeds, item_embeds); acc = e0
// ---------------------------------------------------------------------------
__global__ void k_init(const float* __restrict__ ue, const float* __restrict__ ie,
                       float* __restrict__ e, float* __restrict__ acc,
                       int n_user_elems, int n_total_elems) {
  int i = blockIdx.x * blockDim.x + threadIdx.x;   // float4 index
  int idx4 = i * 4;
  if (idx4 >= n_total_elems) return;
  float4 v;
  if (idx4 < n_user_elems)
    v = ((const float4*)ue)[i];
  else
    v = ((const float4*)ie)[(idx4 - n_user_elems) >> 2];
  ((float4*)e)[i] = v;
  ((float4*)acc)[i] = v;
}

__global__ void k_zero(float* __restrict__ p, int n_elems) {
  int i = blockIdx.x * blockDim.x + threadIdx.x;
  if (i * 4 >= n_elems) return;
  ((float4*)p)[i] = make_float4(0.f, 0.f, 0.f, 0.f);
}

__global__ void k_accum(float* __restrict__ acc, const float* __restrict__ x, int n_elems) {
  int i = blockIdx.x * blockDim.x + threadIdx.x;
  if (i * 4 >= n_elems) return;
  float4 a = ((const float4*)acc)[i];
  float4 b = ((const float4*)x)[i];
  a.x += b.x; a.y += b.y; a.z += b.z; a.w += b.w;
  ((float4*)acc)[i] = a;
}

// ---------------------------------------------------------------------------
// COO SpMM scatter with run-length aggregation.
// One wave32 handles EPW consecutive edges; lane l owns dims 2l, 2l+1.
// The gather of e[c] is a coalesced 256B load per edge. Because the user-side
// half of the edge list is sorted by destination row (avg run ~32), partial
// sums for equal rows are kept in registers and flushed with a single pair of
// native fp32 atomics per run. All row comparisons are wave-uniform, so EXEC
// stays full and control flow is scalar-branch only.
// ---------------------------------------------------------------------------
__global__ void k_scatter(const int* __restrict__ rows, const int* __restrict__ cols,
                          const float* __restrict__ vals,
                          const float* __restrict__ e, float* __restrict__ enext,
                          int n_edges) {
  int  wave = (blockIdx.x * blockDim.x + threadIdx.x) >> 5;
  int  lane = threadIdx.x & 31;
  long base = (long)wave * EPW;
  if (base >= n_edges) return;
  int count = (n_edges - base < EPW) ? (int)(n_edges - base) : EPW;

  // Pull upcoming edge metadata into cache ahead of the wave front
  // (gfx1250 global_prefetch_b8; all lanes hit one line -> one request).
  long pf = base + 256 * EPW;
  if (pf >= n_edges) pf = n_edges - 1;
  __builtin_prefetch(&rows[pf], 0, 0);
  __builtin_prefetch(&cols[pf], 0, 0);
  __builtin_prefetch(&vals[pf], 0, 0);

  float ax = 0.f, ay = 0.f;
  int cur_row = rows[base];

  for (int k = 0; k < count; ++k) {
    int   r = rows[base + k];
    int   c = cols[base + k];
    float v = vals[base + k];
    float2 x = ((const float2*)(e + (size_t)c * EMBED))[lane];

    if (r != cur_row) {                               // wave-uniform branch
      float* dst = enext + (size_t)cur_row * EMBED + lane * 2;
      atomic_add_f32(dst,     ax);
      atomic_add_f32(dst + 1, ay);
      ax = 0.f; ay = 0.f;
      cur_row = r;
    }
    ax += v * x.x;
    ay += v * x.y;
  }
  float* dst = enext + (size_t)cur_row * EMBED + lane * 2;
  atomic_add_f32(dst,     ax);
  atomic_add_f32(dst + 1, ay);
}

// ---------------------------------------------------------------------------
// Finalize: out = 0.25 * acc, tiled 16x16 through V_WMMA_F32_16X16X4_F32.
// Per tile: chain 4 WMMAs, A_s = 0.25 * shifted partial identity (16x4),
// B_s = rows 4s..4s+3 of the tile. fp32-exact (RNE), full EXEC in WMMA branch.
//
// ISA layouts (wave32, lane l):
//   A 16x4 f32 : vgpr j -> A[M = l&15, K = (l>>4)*2 + j]
//   B 4x16 f32 : vgpr j -> B[K = (l>>4)*2 + j, N = l&15]
//   C/D 16x16  : vgpr j -> D[M = j + 8*(l>>4), N = l&15]
// ---------------------------------------------------------------------------
__global__ void k_finalize(float* __restrict__ out, int n_rows) {
  int wave = (blockIdx.x * blockDim.x + threadIdx.x) >> 5;
  int lane = threadIdx.x & 31;
  const int col_tiles = EMBED / 16;                  // 4
  int row_tiles = (n_rows + 15) >> 4;
  if (wave >= row_tiles * col_tiles) return;

  int rt = wave / col_tiles;
  int ct = wave - rt * col_tiles;
  int r0 = rt * 16;
  int c0 = ct * 16;

  int m  = lane & 15;
  int kb = (lane >> 4) * 2;

  if (r0 + 16 <= n_rows) {                           // wave-uniform: EXEC all 1s
    v8f d = {};
#pragma unroll
    for (int s = 0; s < 4; ++s) {
      v2f a, b;
      a.x = (m == 4 * s + kb)     ? 0.25f : 0.0f;
      a.y = (m == 4 * s + kb + 1) ? 0.25f : 0.0f;
      const float* src = out + (size_t)(r0 + 4 * s + kb) * EMBED + c0 + m;
      b.x = src[0];
      b.y = src[EMBED];
      d = __builtin_amdgcn_wmma_f32_16x16x4_f32(
          /*neg_a=*/false, a, /*neg_b=*/false, b,
          /*c_mod=*/(short)0, d, /*reuse_a=*/false, /*reuse_b=*/false);
    }
    int half = lane >> 4;
#pragma unroll
    for (int j = 0; j < 8; ++j)
      out[(size_t)(r0 + j + 8 * half) * EMBED + c0 + m] = d[j];
  } else {
    // Tail rows (n_rows % 16): scalar scale.
    int nr = n_rows - r0;
    for (int t = lane; t < nr * 16; t += 32) {
      int rr = t >> 4, cc = t & 15;
      size_t idx = (size_t)(r0 + rr) * EMBED + c0 + cc;
      out[idx] *= 0.25f;
    }
  }
}

// ---------------------------------------------------------------------------
extern "C" void kernel_launch(void* const* d_in, const int* in_sizes, int n_in,
                              void* d_out, int out_size, void* d_ws, size_t ws_size,
                              hipStream_t stream) {
  const float* user_e = (const float*)d_in[0];
  const float* item_e = (const float*)d_in[1];
  const float* vals   = (const float*)d_in[2];
  const int*   rows   = (const int*)d_in[3];
  const int*   cols   = (const int*)d_in[4];

  int n_user_elems = in_sizes[0];
  int n_item_elems = in_sizes[1];
  int n_edges      = in_sizes[2];
  int n_total      = n_user_elems + n_item_elems;    // == out_size
  int n_rows       = n_total / EMBED;

  float* acc  = (float*)d_out;                       // running layer sum
  float* bufA = (float*)d_ws;                        // ping-pong embedding buffers
  float* bufB = bufA + n_total;                      // 2 x 19.2 MB in ws

  const int blk = 256;
  int n4  = n_total / 4;
  int gb4 = (n4 + blk - 1) / blk;

  k_init<<<gb4, blk, 0, stream>>>(user_e, item_e, bufA, acc, n_user_elems, n_total);

  float* cur = bufA;
  float* nxt = bufB;
  long n_waves   = ((long)n_edges + EPW - 1) / EPW;
  long sc_threads = n_waves * 32;
  int  sc_blocks  = (int)((sc_threads + blk - 1) / blk);

  for (int l = 0; l < N_LAYERS; ++l) {
    k_zero<<<gb4, blk, 0, stream>>>(nxt, n_total);
    k_scatter<<<sc_blocks, blk, 0, stream>>>(rows, cols, vals, cur, nxt, n_edges);
    k_accum<<<gb4, blk, 0, stream>>>(acc, nxt, n_total);
    float* t = cur; cur = nxt; nxt = t;
  }

  int row_tiles = (n_rows + 15) / 16;
  int waves     = row_tiles * (EMBED / 16);
  int fin_blocks = (waves * 32 + blk - 1) / blk;
  k_finalize<<<fin_blocks, blk, 0, stream>>>(acc, n_rows);
}